// GCNGraph_25314537242717
// MI455X (gfx1250) — compile-verified
//
#include <hip/hip_runtime.h>
#include <hip/hip_bf16.h>

#define N_NODES 50000
#define N_EDGES 600000
#define FEAT    128
#define NGRAPH  128
#define NCLASS  10

typedef __attribute__((ext_vector_type(2))) float v2f;
typedef __attribute__((ext_vector_type(8))) float v8f;

// ---------------------------------------------------------------------------
// Degree / normalization
// ---------------------------------------------------------------------------
__global__ __launch_bounds__(256) void deg_init_kernel(float* __restrict__ deg) {
    int i = blockIdx.x * blockDim.x + threadIdx.x;
    if (i < N_NODES) deg[i] = 1.0f;             // self-loop contributes 1
}

__global__ __launch_bounds__(256) void deg_accum_kernel(const int* __restrict__ dst,
                                                        float* __restrict__ deg) {
    int e = blockIdx.x * blockDim.x + threadIdx.x;
    if (e < N_EDGES) atomicAdd(&deg[dst[e]], 1.0f);
}

__global__ __launch_bounds__(256) void dinv_kernel(float* __restrict__ deg) {
    int i = blockIdx.x * blockDim.x + threadIdx.x;
    if (i < N_NODES) deg[i] = rsqrtf(deg[i]);   // deg >= 1 always (self-loop)
}

// ---------------------------------------------------------------------------
// FP32 GEMM via V_WMMA_F32_16X16X4_F32.
// Y[N,128] = A[N,128] @ W[128,128].  One block = 16 rows, 8 waves x 16 cols.
// A-matrix 16x4 f32 layout: lanes 0-15 -> K=khalf..khalf+1, lanes 16-31 ->
// K=khalf+2..; B mirrors with N striped across lanes. C/D: VGPR r holds
// M = r + 8*(lane>=16), N = lane&15 within the wave's 16-col slab.
// ---------------------------------------------------------------------------
__global__ __launch_bounds__(256)
void gemm_wmma_f32_kernel(const float* __restrict__ A,
                          const float* __restrict__ W,
                          float* __restrict__ Y) {
    __shared__ float sA[16][132];               // pad 128->132 to avoid bank conflicts

    const int tid  = threadIdx.x;
    const int wave = tid >> 5;
    const int lane = tid & 31;
    const int row0 = blockIdx.x << 4;           // 16 rows per block

    // Cooperatively stage the 16x128 A tile (2048 floats) into LDS.
    const float4* Av = (const float4*)(A + (size_t)row0 * FEAT);
    for (int f = tid; f < 512; f += 256) {
        float4 v = Av[f];
        int r = f >> 5;                         // 32 float4 per row
        int c = (f & 31) << 2;
        sA[r][c + 0] = v.x; sA[r][c + 1] = v.y;
        sA[r][c + 2] = v.z; sA[r][c + 3] = v.w;
    }
    __syncthreads();

    const int mrow  = lane & 15;
    const int khalf = (lane >> 4) << 1;         // 0 for lanes 0-15, 2 for 16-31
    const int ncol  = (wave << 4) + mrow;       // this wave's output column

    v8f acc = {};
    #pragma unroll
    for (int k = 0; k < 32; ++k) {              // K = 128 = 32 * 4
        const int kb = (k << 2) + khalf;
        v2f a, b;
        a.x = sA[mrow][kb];
        a.y = sA[mrow][kb + 1];
        b.x = W[(size_t)kb       * FEAT + ncol];
        b.y = W[(size_t)(kb + 1) * FEAT + ncol];
        acc = __builtin_amdgcn_wmma_f32_16x16x4_f32(
            /*neg_a=*/false, a, /*neg_b=*/false, b,
            /*c_mod=*/(short)0, acc, /*reuse_a=*/false, /*reuse_b=*/false);
    }

    const int hi = lane >> 4;                   // row half select
    float* out = Y + (size_t)row0 * FEAT + ncol;
    #pragma unroll
    for (int r = 0; r < 8; ++r) {
        out[(size_t)(r + (hi << 3)) * FEAT] = acc[r];
    }
}

// ---------------------------------------------------------------------------
// agg[i] = dinv[i]^2 * hw[i] + bias          (self-loop term + bias, no atomics)
// ---------------------------------------------------------------------------
__global__ __launch_bounds__(256)
void init_agg_kernel(const float* __restrict__ hw, const float* __restrict__ dinv,
                     const float* __restrict__ bias, float* __restrict__ agg) {
    int t = blockIdx.x * blockDim.x + threadIdx.x;          // N * 32 threads
    if (t >= N_NODES * 32) return;
    int i  = t >> 5;
    int f4 = t & 31;
    float di  = dinv[i];
    float nrm = di * di;
    float4 m = ((const float4*)(hw  + (size_t)i * FEAT))[f4];
    float4 b = ((const float4*)bias)[f4];
    float4 o;
    o.x = nrm * m.x + b.x; o.y = nrm * m.y + b.y;
    o.z = nrm * m.z + b.z; o.w = nrm * m.w + b.w;
    ((float4*)(agg + (size_t)i * FEAT))[f4] = o;
}

// ---------------------------------------------------------------------------
// agg[dst] += dinv[src]*dinv[dst] * hw[src]   (edge scatter, 4 floats/thread)
// ---------------------------------------------------------------------------
__global__ __launch_bounds__(256)
void scatter_kernel(const float* __restrict__ hw,
                    const int* __restrict__ src, const int* __restrict__ dst,
                    const float* __restrict__ dinv, float* __restrict__ agg) {
    long long t = (long long)blockIdx.x * blockDim.x + threadIdx.x;
    if (t >= (long long)N_EDGES * 32) return;
    int e  = (int)(t >> 5);
    int f4 = (int)(t & 31);
    int s = src[e], d = dst[e];
    float nrm = dinv[s] * dinv[d];
    float4 m = ((const float4*)(hw + (size_t)s * FEAT))[f4];
    float* o = agg + (size_t)d * FEAT + (f4 << 2);
    atomicAdd(o + 0, nrm * m.x);
    atomicAdd(o + 1, nrm * m.y);
    atomicAdd(o + 2, nrm * m.z);
    atomicAdd(o + 3, nrm * m.w);
}

__global__ __launch_bounds__(256) void relu_kernel(float* __restrict__ h) {
    long long t = (long long)blockIdx.x * blockDim.x + threadIdx.x;
    if (t < (long long)N_NODES * FEAT) h[t] = fmaxf(h[t], 0.0f);
}

// ---------------------------------------------------------------------------
// Global mean pool + classifier
// ---------------------------------------------------------------------------
__global__ __launch_bounds__(256) void pool_zero_kernel(float* __restrict__ pooled,
                                                        float* __restrict__ cnt) {
    int t = blockIdx.x * blockDim.x + threadIdx.x;
    if (t < NGRAPH * FEAT) pooled[t] = 0.0f;
    if (t < NGRAPH) cnt[t] = 0.0f;
}

__global__ __launch_bounds__(256)
void pool_kernel(const float* __restrict__ h, const int* __restrict__ batch,
                 float* __restrict__ pooled, float* __restrict__ cnt) {
    int t = blockIdx.x * blockDim.x + threadIdx.x;          // N * 32 threads
    if (t >= N_NODES * 32) return;
    int i  = t >> 5;
    int f4 = t & 31;
    int g  = batch[i];
    float4 v = ((const float4*)(h + (size_t)i * FEAT))[f4];
    float* o = pooled + (size_t)g * FEAT + (f4 << 2);
    atomicAdd(o + 0, v.x);
    atomicAdd(o + 1, v.y);
    atomicAdd(o + 2, v.z);
    atomicAdd(o + 3, v.w);
    if (f4 == 0) atomicAdd(&cnt[g], 1.0f);
}

__global__ __launch_bounds__(256)
void classifier_kernel(const float* __restrict__ pooled, const float* __restrict__ cnt,
                       const float* __restrict__ Wl, const float* __restrict__ bl,
                       float* __restrict__ out) {
    int t = blockIdx.x * blockDim.x + threadIdx.x;          // G * C threads
    if (t >= NGRAPH * NCLASS) return;
    int g = t / NCLASS;
    int c = t % NCLASS;
    float s = 0.0f;
    #pragma unroll 8
    for (int k = 0; k < FEAT; ++k)
        s += pooled[(size_t)g * FEAT + k] * Wl[(size_t)k * NCLASS + c];
    out[t] = s / fmaxf(cnt[g], 1.0f) + bl[c];
}

// ---------------------------------------------------------------------------
// Host-side launcher
// ---------------------------------------------------------------------------
extern "C" void kernel_launch(void* const* d_in, const int* in_sizes, int n_in,
                              void* d_out, int out_size, void* d_ws, size_t ws_size,
                              hipStream_t stream) {
    // setup_inputs order: x, W1, b1, W2, b2, W3, b3, Wl, bl, edge_index, batch
    const float* x  = (const float*)d_in[0];
    const float* W1 = (const float*)d_in[1];
    const float* b1 = (const float*)d_in[2];
    const float* W2 = (const float*)d_in[3];
    const float* b2 = (const float*)d_in[4];
    const float* W3 = (const float*)d_in[5];
    const float* b3 = (const float*)d_in[6];
    const float* Wl = (const float*)d_in[7];
    const float* bl = (const float*)d_in[8];
    const int*   ei = (const int*)d_in[9];
    const int*   batch = (const int*)d_in[10];
    float* out = (float*)d_out;

    const int* src = ei;                 // edge_index[0]
    const int* dst = ei + N_EDGES;       // edge_index[1]

    // Workspace layout (floats, 256-float aligned chunks)
    float* ws = (float*)d_ws;
    const size_t NP   = 50176;                       // N rounded up
    const size_t NF   = (size_t)N_NODES * FEAT;      // 6,400,000
    float* dinv   = ws;                              // [N]     (deg -> dinv in place)
    float* hw     = ws + NP;                         // [N,128] gemm output
    float* bufA   = hw + NF;                         // [N,128]
    float* bufB   = bufA + NF;                       // [N,128]
    float* pooled = bufB + NF;                       // [G,128]
    float* cnt    = pooled + (size_t)NGRAPH * FEAT;  // [G]

    const int T = 256;
    const int gN    = (N_NODES + T - 1) / T;               // 196
    const int gE    = (N_EDGES + T - 1) / T;               // 2344
    const int gGemm = N_NODES / 16;                        // 3125 (N % 16 == 0)
    const int gN32  = (N_NODES * 32) / T;                  // 6250
    const int gE32  = (N_EDGES * 32) / T;                  // 75000
    const int gNF   = (int)(NF / T);                       // 25000

    // --- degree / symmetric norm ---
    deg_init_kernel <<<gN, T, 0, stream>>>(dinv);
    deg_accum_kernel<<<gE, T, 0, stream>>>(dst, dinv);
    dinv_kernel     <<<gN, T, 0, stream>>>(dinv);

    // --- layer 1: x -> bufA ---
    gemm_wmma_f32_kernel<<<gGemm, T, 0, stream>>>(x, W1, hw);
    init_agg_kernel     <<<gN32, T, 0, stream>>>(hw, dinv, b1, bufA);
    scatter_kernel      <<<gE32, T, 0, stream>>>(hw, src, dst, dinv, bufA);
    relu_kernel         <<<gNF,  T, 0, stream>>>(bufA);

    // --- layer 2: bufA -> bufB ---
    gemm_wmma_f32_kernel<<<gGemm, T, 0, stream>>>(bufA, W2, hw);
    init_agg_kernel     <<<gN32, T, 0, stream>>>(hw, dinv, b2, bufB);
    scatter_kernel      <<<gE32, T, 0, stream>>>(hw, src, dst, dinv, bufB);
    relu_kernel         <<<gNF,  T, 0, stream>>>(bufB);

    // --- layer 3: bufB -> bufA ---
    gemm_wmma_f32_kernel<<<gGemm, T, 0, stream>>>(bufB, W3, hw);
    init_agg_kernel     <<<gN32, T, 0, stream>>>(hw, dinv, b3, bufA);
    scatter_kernel      <<<gE32, T, 0, stream>>>(hw, src, dst, dinv, bufA);
    relu_kernel         <<<gNF,  T, 0, stream>>>(bufA);

    // --- global mean pool + classifier ---
    pool_zero_kernel <<<(NGRAPH * FEAT + T - 1) / T, T, 0, stream>>>(pooled, cnt);
    pool_kernel      <<<gN32, T, 0, stream>>>(bufA, batch, pooled, cnt);
    classifier_kernel<<<(NGRAPH * NCLASS + T - 1) / T, T, 0, stream>>>(pooled, cnt, Wl, bl, out);
}